// ASSM_4964982194541
// MI455X (gfx1250) — compile-verified
//
#include <hip/hip_runtime.h>
#include <cstddef>

#define Bsz   4
#define Lsz   4096
#define DMsz  192
#define Nsz   16
#define Rsz   12
#define Tsz   64
#define H3sz  64
#define CROWS 44           // R + 2N
#define HSTR  66           // LDS stride for 64-deep columns (66 mod 64 == 2 -> conflict free)
#define DSTR  46           // LDS stride for 44-deep columns

typedef __attribute__((ext_vector_type(2))) float v2f;
typedef __attribute__((ext_vector_type(8))) float v8f;

// ---- CDNA5 fp32 WMMA 16x16x4:  D = A(16x4) * B(4x16) + C ----------------
__device__ __forceinline__ v8f wmma_f32_k4(v2f a, v2f b, v8f c) {
#if __has_builtin(__builtin_amdgcn_wmma_f32_16x16x4_f32)
  return __builtin_amdgcn_wmma_f32_16x16x4_f32(
      /*neg_a=*/false, a, /*neg_b=*/false, b,
      /*c_mod=*/(short)0, c, /*reuse_a=*/false, /*reuse_b=*/false);
#else
  // degenerate fallback so the TU still compiles if the builtin is absent
  c[0] += a.x * b.x;
  return c;
#endif
}

// =========================================================================
// Kernel 0: M_s = emb_s (64x12) @ token_s_w (12x16)  -> ws (2 x 64 x 16)
// =========================================================================
__global__ void assm_mk(const float* __restrict__ emb0, const float* __restrict__ tw0,
                        const float* __restrict__ emb1, const float* __restrict__ tw1,
                        float* __restrict__ Mws) {
  int idx = blockIdx.x * blockDim.x + threadIdx.x;
  if (idx >= 2 * Tsz * Nsz) return;
  int s = idx >> 10;
  int t = (idx >> 4) & 63;
  int n = idx & 15;
  const float* emb = s ? emb1 : emb0;
  const float* tw  = s ? tw1 : tw0;
  float acc = 0.f;
  for (int r = 0; r < Rsz; ++r) acc += emb[t * Rsz + r] * tw[r * Nsz + n];
  Mws[idx] = acc;
}

// =========================================================================
// Kernel 1 (per stream): routing + projections + gumbel argmax + delta/B/C
// One wave per 16-token tile, all GEMMs via v_wmma_f32_16x16x4_f32.
// block = 128 threads (4 waves) => 64 tokens / block; grid = (L/64, B)
// =========================================================================
__global__ __launch_bounds__(128)
void assm_front(const float* __restrict__ x,      // (B,L,DM)
                const float* __restrict__ u,      // (B,L,T)
                const float* __restrict__ Mw,     // (T,N) = 64x16
                const float* __restrict__ w1,     // (H3,DM)
                const float* __restrict__ b1,     // (H3)
                const float* __restrict__ w2,     // (T,H3)
                const float* __restrict__ b2,     // (T)
                const float* __restrict__ xproj,  // (44,DM)
                const float* __restrict__ dtw,    // (DM,R)
                const float* __restrict__ dtb,    // (DM)
                float* __restrict__ delta_out,    // (B,DM,L)
                float* __restrict__ Bout,         // (B,L,N)
                float* __restrict__ Cout)         // (B,L,N)
{
  const int lane = threadIdx.x & 31;
  const int wv   = threadIdx.x >> 5;
  const int b    = blockIdx.y;
  const int t0   = (blockIdx.x * 4 + wv) * 16;   // first token of this wave's tile
  const int mn   = lane & 15;                    // A-row m == B-col (token) n
  const int hi   = lane >> 4;
  const int kb   = 2 * hi;                       // K sub-offset per ISA layout

  __shared__ float Ht  [4][16 * HSTR];           // h   (token-major, 64 deep)
  __shared__ float DBLt[4][16 * DSTR];           // dbl (token-major, 44 deep)
  __shared__ float LOGt[4][16 * HSTR];           // logits (token-major, 64 deep)
  float* ht  = Ht[wv];
  float* dbl = DBLt[wv];
  float* lg  = LOGt[wv];

  const float* xb = x + ((size_t)b * Lsz + t0) * DMsz;   // 16 contiguous token rows
  __builtin_prefetch(xb + 16 * DMsz, 0, 1);              // speculative: next tile

  // ---- dbl = xproj @ x_tile : 3 row tiles of 16 (rows 44..47 masked) ----
  for (int rt = 0; rt < 3; ++rt) {
    const int  row = rt * 16 + mn;
    const bool rok = row < CROWS;
    const float* wrow = xproj + (size_t)row * DMsz;
    v8f acc = {};
    for (int k0 = 0; k0 < DMsz; k0 += 4) {
      v2f a; a.x = rok ? wrow[k0 + kb] : 0.f;  a.y = rok ? wrow[k0 + kb + 1] : 0.f;
      const float* xc = xb + mn * DMsz + k0 + kb;
      v2f bf; bf.x = xc[0]; bf.y = xc[1];
      acc = wmma_f32_k4(a, bf, acc);
    }
    for (int v = 0; v < 8; ++v) {
      int r = rt * 16 + v + 8 * hi;
      if (r < CROWS) dbl[mn * DSTR + r] = acc[v];
    }
  }

  // ---- h = gelu(w1 @ x + b1) : 4 row tiles ------------------------------
  for (int rt = 0; rt < 4; ++rt) {
    const float* wrow = w1 + (size_t)(rt * 16 + mn) * DMsz;
    v8f acc = {};
    for (int k0 = 0; k0 < DMsz; k0 += 4) {
      v2f a; a.x = wrow[k0 + kb]; a.y = wrow[k0 + kb + 1];
      const float* xc = xb + mn * DMsz + k0 + kb;
      v2f bf; bf.x = xc[0]; bf.y = xc[1];
      acc = wmma_f32_k4(a, bf, acc);
    }
    for (int v = 0; v < 8; ++v) {
      int r = rt * 16 + v + 8 * hi;
      float hv = acc[v] + b1[r];
      hv = 0.5f * hv * (1.0f + erff(hv * 0.70710678118f));  // exact gelu
      ht[mn * HSTR + r] = hv;
    }
  }

  // ---- logits = w2 @ h + b2 : 4 row tiles, K=64 from LDS ----------------
  for (int rt = 0; rt < 4; ++rt) {
    const float* wrow = w2 + (size_t)(rt * 16 + mn) * H3sz;
    v8f acc = {};
    for (int k0 = 0; k0 < H3sz; k0 += 4) {
      v2f a; a.x = wrow[k0 + kb]; a.y = wrow[k0 + kb + 1];
      const float* hc = ht + mn * HSTR + k0 + kb;
      v2f bf; bf.x = hc[0]; bf.y = hc[1];
      acc = wmma_f32_k4(a, bf, acc);
    }
    for (int v = 0; v < 8; ++v) {
      int r = rt * 16 + v + 8 * hi;
      lg[mn * HSTR + r] = acc[v] + b2[r];
    }
  }

  // ---- dt = dtw @ dbl[0:12], delta = softplus(dt + dtb) : 12 row tiles --
  for (int rt = 0; rt < 12; ++rt) {
    const float* wrow = dtw + (size_t)(rt * 16 + mn) * Rsz;
    v8f acc = {};
    for (int k0 = 0; k0 < 12; k0 += 4) {
      v2f a; a.x = wrow[k0 + kb]; a.y = wrow[k0 + kb + 1];
      const float* dc = dbl + mn * DSTR + k0 + kb;
      v2f bf; bf.x = dc[0]; bf.y = dc[1];
      acc = wmma_f32_k4(a, bf, acc);
    }
    for (int v = 0; v < 8; ++v) {
      int d  = rt * 16 + v + 8 * hi;
      float dv = acc[v] + dtb[d];
      dv = (dv > 20.0f) ? dv : log1pf(expf(dv));           // softplus
      delta_out[((size_t)b * DMsz + d) * Lsz + (t0 + mn)] = dv;
    }
  }

  // ---- per-token gumbel argmax + B/C outputs (lanes 0..15) --------------
  if (lane < 16) {
    const int n = lane;
    const int l = t0 + n;
    const float* urow = u + ((size_t)b * Lsz + l) * Tsz;
    float best = -3.0e38f; int bi = 0;
    for (int t = 0; t < Tsz; ++t) {
      float g = -logf(-logf(urow[t]));                     // gumbel noise
      float v = lg[n * HSTR + t] + g;                      // log_softmax shift is argmax-invariant
      if (v > best) { best = v; bi = t; }
    }
    const float* mrow = Mw + bi * Nsz;                     // prompt row gather
    float* bo = Bout + (((size_t)b * Lsz) + l) * Nsz;
    float* co = Cout + (((size_t)b * Lsz) + l) * Nsz;
    for (int nn = 0; nn < Nsz; ++nn) {
      bo[nn] = dbl[n * DSTR + (Rsz + nn)];
      co[nn] = dbl[n * DSTR + (Rsz + Nsz + nn)] + mrow[nn];
    }
  }
}

// =========================================================================
// Kernel 2 (per stream): selective scan. 16 lanes = 16 states of one d.
// block = 128 (4 waves = 8 channels); grid = (DM/8, B)
// =========================================================================
__global__ __launch_bounds__(128)
void assm_scan(const float* __restrict__ delta,  // (B,DM,L)
               const float* __restrict__ Bw,     // (B,L,N)
               const float* __restrict__ Cw,     // (B,L,N) — cross stream!
               const float* __restrict__ x,      // (B,L,DM) == u
               const float* __restrict__ Alog,   // (DM,N)
               const float* __restrict__ Dv,     // (DM)
               float* __restrict__ yout)         // (B,DM,L) pre-LN
{
  const int lane = threadIdx.x & 31;
  const int wv   = threadIdx.x >> 5;
  const int n    = lane & 15;
  const int b    = blockIdx.y;
  const int d    = blockIdx.x * 8 + wv * 2 + (lane >> 4);

  const float A  = -expf(Alog[d * Nsz + n]);
  const float Dd = Dv[d];
  const float* drow = delta + ((size_t)b * DMsz + d) * Lsz;
  const float* brow = Bw + ((size_t)b * Lsz) * Nsz + n;
  const float* crow = Cw + ((size_t)b * Lsz) * Nsz + n;
  const float* xrow = x  + ((size_t)b * Lsz) * DMsz + d;
  float* yrow = yout + ((size_t)b * DMsz + d) * Lsz;

  float h = 0.f;
  for (int l = 0; l < Lsz; ++l) {
    float dl = drow[l];
    float uv = xrow[(size_t)l * DMsz];
    float bv = brow[(size_t)l * Nsz];
    float cv = crow[(size_t)l * Nsz];
    h = expf(dl * A) * h + dl * uv * bv;
    float p = h * cv;                                // reduce over n (16 lanes)
    p += __shfl_xor(p, 1, 32);
    p += __shfl_xor(p, 2, 32);
    p += __shfl_xor(p, 4, 32);
    p += __shfl_xor(p, 8, 32);
    if (n == 0) yrow[l] = p + Dd * uv;
  }
}

// =========================================================================
// Kernel 3 (per stream): in-place LayerNorm over DM. 1 wave per token.
// block = 128 (4 tokens); grid = B*L/4
// =========================================================================
__global__ __launch_bounds__(128)
void assm_ln(float* __restrict__ y,               // (B,DM,L) in d_out
             const float* __restrict__ g,
             const float* __restrict__ beta)
{
  const int lane = threadIdx.x & 31;
  const int wv   = threadIdx.x >> 5;
  const int tok  = blockIdx.x * 4 + wv;           // 0 .. B*L-1
  const int b    = tok / Lsz;
  const int l    = tok - b * Lsz;
  float* base = y + ((size_t)b * DMsz) * Lsz + l;

  float vals[6];
  float s = 0.f;
  for (int i = 0; i < 6; ++i) {
    vals[i] = base[(size_t)(lane + 32 * i) * Lsz];
    s += vals[i];
  }
  for (int off = 16; off; off >>= 1) s += __shfl_xor(s, off, 32);
  const float mean = s * (1.0f / DMsz);
  float vs = 0.f;
  for (int i = 0; i < 6; ++i) { float t = vals[i] - mean; vs += t * t; }
  for (int off = 16; off; off >>= 1) vs += __shfl_xor(vs, off, 32);
  const float inv = rsqrtf(vs * (1.0f / DMsz) + 1e-5f);
  for (int i = 0; i < 6; ++i) {
    int dd = lane + 32 * i;
    base[(size_t)dd * Lsz] = (vals[i] - mean) * inv * g[dd] + beta[dd];
  }
}

// =========================================================================
extern "C" void kernel_launch(void* const* d_in, const int* in_sizes, int n_in,
                              void* d_out, int out_size, void* d_ws, size_t ws_size,
                              hipStream_t stream) {
  const float* x_rgb   = (const float*)d_in[0];
  const float* x_e     = (const float*)d_in[1];
  const float* tw_rgb  = (const float*)d_in[2];
  const float* tw_e    = (const float*)d_in[3];
  const float* u_rgb   = (const float*)d_in[4];
  const float* u_e     = (const float*)d_in[5];
  const float* emb_rgb = (const float*)d_in[6];
  const float* emb_e   = (const float*)d_in[7];
  const float* rw1     = (const float*)d_in[8];
  const float* rb1     = (const float*)d_in[9];
  const float* rw2     = (const float*)d_in[10];
  const float* rb2     = (const float*)d_in[11];
  const float* ew1     = (const float*)d_in[12];
  const float* eb1     = (const float*)d_in[13];
  const float* ew2     = (const float*)d_in[14];
  const float* eb2     = (const float*)d_in[15];
  const float* xp_rgb  = (const float*)d_in[16];
  const float* xp_e    = (const float*)d_in[17];
  const float* dtw_rgb = (const float*)d_in[18];
  const float* dtb_rgb = (const float*)d_in[19];
  const float* dtw_e   = (const float*)d_in[20];
  const float* dtb_e   = (const float*)d_in[21];
  const float* Al_rgb  = (const float*)d_in[22];
  const float* Al_e    = (const float*)d_in[23];
  const float* D_rgb   = (const float*)d_in[24];
  const float* D_e     = (const float*)d_in[25];
  const float* ln1_g   = (const float*)d_in[26];
  const float* ln1_b   = (const float*)d_in[27];
  const float* ln2_g   = (const float*)d_in[28];
  const float* ln2_b   = (const float*)d_in[29];

  const size_t BDL = (size_t)Bsz * DMsz * Lsz;   // 3,145,728
  const size_t BLN = (size_t)Bsz * Lsz * Nsz;    //   262,144

  float* ws      = (float*)d_ws;
  float* Mws     = ws;                           // 2 * 1024
  float* deltaWs = Mws + 2048;                   // 2 * BDL
  float* BWs     = deltaWs + 2 * BDL;            // 2 * BLN
  float* CWs     = BWs + 2 * BLN;                // 2 * BLN

  float* out = (float*)d_out;

  // prompt matrices
  assm_mk<<<dim3(16), 128, 0, stream>>>(emb_rgb, tw_rgb, emb_e, tw_e, Mws);

  // front-end (WMMA GEMMs) per stream
  dim3 fg(Lsz / 64, Bsz);
  assm_front<<<fg, 128, 0, stream>>>(x_rgb, u_rgb, Mws,        rw1, rb1, rw2, rb2,
                                     xp_rgb, dtw_rgb, dtb_rgb,
                                     deltaWs,        BWs,        CWs);
  assm_front<<<fg, 128, 0, stream>>>(x_e,   u_e,   Mws + 1024, ew1, eb1, ew2, eb2,
                                     xp_e,  dtw_e,  dtb_e,
                                     deltaWs + BDL,  BWs + BLN,  CWs + BLN);

  // selective scan (cross-stream C) per stream
  dim3 sg(DMsz / 8, Bsz);
  assm_scan<<<sg, 128, 0, stream>>>(deltaWs,       BWs,       CWs + BLN, x_rgb,
                                    Al_rgb, D_rgb, out);
  assm_scan<<<sg, 128, 0, stream>>>(deltaWs + BDL, BWs + BLN, CWs,       x_e,
                                    Al_e,   D_e,   out + BDL);

  // layernorm in place
  dim3 lg2((Bsz * Lsz) / 4);
  assm_ln<<<lg2, 128, 0, stream>>>(out,       ln1_g, ln1_b);
  assm_ln<<<lg2, 128, 0, stream>>>(out + BDL, ln2_g, ln2_b);
}